// GCN_43688407334924
// MI455X (gfx1250) — compile-verified
//
#include <hip/hip_runtime.h>
#include <hip/hip_bf16.h>

typedef __attribute__((ext_vector_type(2))) float v2f;
typedef __attribute__((ext_vector_type(8))) float v8f;

// ---------------------------------------------------------------------------
// Degree / normalization kernels
// ---------------------------------------------------------------------------
__global__ void deg_init_kernel(float* __restrict__ deg, int N) {
    int i = blockIdx.x * blockDim.x + threadIdx.x;
    if (i < N) deg[i] = 1.0f;   // self-loop weight
}

__global__ void deg_count_kernel(const int* __restrict__ ei, float* __restrict__ deg, int E) {
    long long gid = (long long)blockIdx.x * blockDim.x + threadIdx.x;
    long long stride = (long long)gridDim.x * blockDim.x;
    for (; gid < E; gid += stride) {
        int d = ei[E + gid];    // dst row
        atomicAdd(&deg[d], 1.0f);
    }
}

__global__ void make_dinv_kernel(float* __restrict__ deg, int N) {
    int i = blockIdx.x * blockDim.x + threadIdx.x;
    if (i < N) deg[i] = rsqrtf(deg[i]);   // in-place deg -> dinv
}

__global__ void make_enorm_kernel(const int* __restrict__ ei, const float* __restrict__ dinv,
                                  float* __restrict__ enorm, int E) {
    long long gid = (long long)blockIdx.x * blockDim.x + threadIdx.x;
    long long stride = (long long)gridDim.x * blockDim.x;
    for (; gid < E; gid += stride) {
        int s = ei[gid];
        int d = ei[E + gid];
        enorm[gid] = dinv[s] * dinv[d];
    }
}

// ---------------------------------------------------------------------------
// WMMA f32 GEMM:  H[M,Fo] = (relu_in ? relu(A) : A)[M,K] @ B[K,Fo]
//
// Block = 256 threads = 8 waves. blockIdx.x -> 8 consecutive 16-row M tiles
// (one per wave), blockIdx.y -> one 16-column N tile shared by the block.
//
// The 16-column B panel (K x 16 floats, <= 16 KB) is staged in LDS once per
// block, pair-major: smem[(p*16 + col)*2 + parity] = B[2p+parity][col].
// Each lane's B fragment is then one conflict-free ds_load_b64 (a full wave
// reads 256 consecutive bytes = all 64 banks). Inner loop has no divergence.
//
// A frag (16x4 f32): lane<16 -> K = k0+{0,1}; lane>=16 -> K = k0+{2,3}
// C/D: VGPR r -> M = m0 + r + 8*(lane>>4), N = n0 + (lane&15)
// ---------------------------------------------------------------------------
__global__ void __launch_bounds__(256)
gemm_wmma_f32_kernel(const float* __restrict__ A, const float* __restrict__ B,
                     float* __restrict__ H, int M, int K, int Fo, int relu_in) {
    __shared__ float smem[256 * 16];            // max K(256) x 16 panel, 16 KB

    const int tid  = threadIdx.x;
    const int wave = tid >> 5;                  // 0..7
    const int lane = tid & 31;
    const int lr   = lane & 15;
    const int half = lane >> 4;

    const int tileM  = blockIdx.x * 8 + wave;   // wave-uniform
    const bool activeM = (tileM * 16 < M);      // OOB waves still hit the barrier
    const int m0  = activeM ? tileM * 16 : 0;   // redundant-but-valid tile for OOB waves
    const int n0  = blockIdx.y * 16;
    const int col = n0 + lr;

    // ---- stage B panel into LDS (pair-major float2 layout) ----
    for (int idx = tid; idx < K * 16; idx += 256) {
        int k = idx >> 4;
        int c = idx & 15;
        int cc = n0 + c;
        float v = (cc < Fo) ? B[(size_t)k * Fo + cc] : 0.0f;
        smem[((k >> 1) * 16 + c) * 2 + (k & 1)] = v;
    }
    __syncthreads();

    const float* __restrict__ arow = A + (size_t)(m0 + lr) * K;
    const v2f* __restrict__ bpanel = (const v2f*)smem;   // indexed by pair*16 + col

    v8f c = {};
    #pragma unroll 4
    for (int k0 = 0; k0 < K; k0 += 4) {
        const int ka = k0 + 2 * half;           // lane's starting K
        v2f a;
        a.x = arow[ka];
        a.y = arow[ka + 1];
        if (relu_in) { a.x = fmaxf(a.x, 0.0f); a.y = fmaxf(a.y, 0.0f); }

        v2f b = bpanel[((k0 >> 1) + half) * 16 + lr];   // ds_load_b64, conflict-free

        // 8 args: (neg_a, A, neg_b, B, c_mod, C, reuse_a, reuse_b)
        c = __builtin_amdgcn_wmma_f32_16x16x4_f32(false, a, false, b, (short)0, c, false, false);
    }

    if (activeM && col < Fo) {
        #pragma unroll
        for (int r = 0; r < 8; ++r) {
            H[(size_t)(m0 + r + 8 * half) * Fo + col] = c[r];
        }
    }
}

// ---------------------------------------------------------------------------
// agg[i,f] = dinv[i]^2 * h[i,f] + bias[f]   (self-loop term + bias, pre-scatter init)
// ---------------------------------------------------------------------------
__global__ void init_agg_kernel(const float* __restrict__ h, const float* __restrict__ dinv,
                                const float* __restrict__ bias, float* __restrict__ agg,
                                int N, int Fo) {
    long long total = (long long)N * Fo;
    long long gid = (long long)blockIdx.x * blockDim.x + threadIdx.x;
    long long stride = (long long)gridDim.x * blockDim.x;
    for (; gid < total; gid += stride) {
        int i = (int)(gid / Fo);
        int f = (int)(gid - (long long)i * Fo);
        float di = dinv[i];
        agg[gid] = di * di * h[gid] + bias[f];
    }
}

// ---------------------------------------------------------------------------
// Edge scatter:  agg[dst] += enorm[e] * h[src]   (float4 vectorized, Fo % 4 == 0,
// chunks per row = 1<<l2c must equal Fo/4)
// ---------------------------------------------------------------------------
__global__ void scatter_vec4_kernel(const int* __restrict__ ei, const float* __restrict__ enorm,
                                    const float* __restrict__ h, float* __restrict__ agg,
                                    int E, int Fo, int l2c) {
    long long total = (long long)E << l2c;
    long long gid = (long long)blockIdx.x * blockDim.x + threadIdx.x;
    long long stride = (long long)gridDim.x * blockDim.x;
    const int cmask = (1 << l2c) - 1;
    for (; gid < total; gid += stride) {
        int e = (int)(gid >> l2c);
        int f = (int)((gid & cmask) << 2);
        int s = ei[e];
        int d = ei[E + e];
        float w = enorm[e];
        const float4 v = *(const float4*)(h + (size_t)s * Fo + f);
        float* ap = agg + (size_t)d * Fo + f;
        atomicAdd(ap + 0, w * v.x);
        atomicAdd(ap + 1, w * v.y);
        atomicAdd(ap + 2, w * v.z);
        atomicAdd(ap + 3, w * v.w);
    }
}

// Scalar variant (one thread per edge, loops over Fo) for Fo not divisible by 4.
__global__ void scatter_scalar_kernel(const int* __restrict__ ei, const float* __restrict__ enorm,
                                      const float* __restrict__ h, float* __restrict__ agg,
                                      int E, int Fo) {
    long long gid = (long long)blockIdx.x * blockDim.x + threadIdx.x;
    long long stride = (long long)gridDim.x * blockDim.x;
    for (; gid < E; gid += stride) {
        int s = ei[gid];
        int d = ei[E + gid];
        float w = enorm[gid];
        const float* hp = h + (size_t)s * Fo;
        float* ap = agg + (size_t)d * Fo;
        for (int f = 0; f < Fo; ++f) atomicAdd(ap + f, w * hp[f]);
    }
}

// ---------------------------------------------------------------------------
// Row-wise log_softmax over C classes (one thread per row; C == 10)
// ---------------------------------------------------------------------------
__global__ void log_softmax_kernel(const float* __restrict__ in, float* __restrict__ out,
                                   int N, int C) {
    int i = blockIdx.x * blockDim.x + threadIdx.x;
    if (i >= N) return;
    const float* r = in + (size_t)i * C;
    float m = r[0];
    for (int c = 1; c < C; ++c) m = fmaxf(m, r[c]);
    float s = 0.0f;
    for (int c = 0; c < C; ++c) s += __expf(r[c] - m);
    float lse = m + __logf(s);
    float* o = out + (size_t)i * C;
    for (int c = 0; c < C; ++c) o[c] = r[c] - lse;
}

// ---------------------------------------------------------------------------
// Host orchestration
// ---------------------------------------------------------------------------
static inline size_t align256(size_t x) { return (x + 255) & ~(size_t)255; }

extern "C" void kernel_launch(void* const* d_in, const int* in_sizes, int n_in,
                              void* d_out, int out_size, void* d_ws, size_t ws_size,
                              hipStream_t stream) {
    const float* x  = (const float*)d_in[0];
    const int*   ei = (const int*)d_in[1];
    const float* W1 = (const float*)d_in[2]; const float* b1 = (const float*)d_in[3];
    const float* W2 = (const float*)d_in[4]; const float* b2 = (const float*)d_in[5];
    const float* W3 = (const float*)d_in[6]; const float* b3 = (const float*)d_in[7];
    const float* W4 = (const float*)d_in[8]; const float* b4 = (const float*)d_in[9];
    float* out = (float*)d_out;

    const int IN_F = 256, H1 = 128, H2 = 256, NCLS = 10;
    const int N = in_sizes[0] / IN_F;     // 50000
    const int E = in_sizes[1] / 2;        // 1600000

    // Workspace carve-out (256-byte aligned)
    char* ws = (char*)d_ws;
    size_t off = 0;
    float* dinv  = (float*)(ws + off); off += align256((size_t)N * 4);
    float* enorm = (float*)(ws + off); off += align256((size_t)E * 4);
    float* bufH  = (float*)(ws + off); off += align256((size_t)N * 256 * 4);  // gemm output, widest 256
    float* buf0  = (float*)(ws + off); off += align256((size_t)N * 128 * 4);  // agg1 / agg3 (width 128)
    float* buf1  = (float*)(ws + off); off += align256((size_t)N * 256 * 4);  // agg2 (256) / agg4 (10)
    (void)ws_size; (void)n_in; (void)out_size;

    const int TB = 256;
    const int SCATTER_BLOCKS = 32768;
    const int EW_BLOCKS = 16384;

    // --- normalization precompute ---
    deg_init_kernel<<<(N + TB - 1) / TB, TB, 0, stream>>>(dinv, N);
    deg_count_kernel<<<SCATTER_BLOCKS, TB, 0, stream>>>(ei, dinv, E);
    make_dinv_kernel<<<(N + TB - 1) / TB, TB, 0, stream>>>(dinv, N);
    make_enorm_kernel<<<SCATTER_BLOCKS, TB, 0, stream>>>(ei, dinv, enorm, E);

    const int mtiles = (N + 15) / 16;             // 3125
    const int gx = (mtiles + 7) / 8;              // 8 waves (tiles) per block

    // --- layer 1: x[256] -> h[128] ---
    gemm_wmma_f32_kernel<<<dim3(gx, H1 / 16), TB, 0, stream>>>(x, W1, bufH, N, IN_F, H1, 0);
    init_agg_kernel<<<EW_BLOCKS, TB, 0, stream>>>(bufH, dinv, b1, buf0, N, H1);
    scatter_vec4_kernel<<<SCATTER_BLOCKS, TB, 0, stream>>>(ei, enorm, bufH, buf0, E, H1, 5);

    // --- layer 2: relu(agg1)[128] -> h[256] ---
    gemm_wmma_f32_kernel<<<dim3(gx, H2 / 16), TB, 0, stream>>>(buf0, W2, bufH, N, H1, H2, 1);
    init_agg_kernel<<<EW_BLOCKS, TB, 0, stream>>>(bufH, dinv, b2, buf1, N, H2);
    scatter_vec4_kernel<<<SCATTER_BLOCKS, TB, 0, stream>>>(ei, enorm, bufH, buf1, E, H2, 6);

    // --- layer 3: relu(agg2)[256] -> h[128] ---
    gemm_wmma_f32_kernel<<<dim3(gx, H1 / 16), TB, 0, stream>>>(buf1, W3, bufH, N, H2, H1, 1);
    init_agg_kernel<<<EW_BLOCKS, TB, 0, stream>>>(bufH, dinv, b3, buf0, N, H1);
    scatter_vec4_kernel<<<SCATTER_BLOCKS, TB, 0, stream>>>(ei, enorm, bufH, buf0, E, H1, 5);

    // --- layer 4: relu(agg3)[128] -> h[10] ---
    gemm_wmma_f32_kernel<<<dim3(gx, 1), TB, 0, stream>>>(buf0, W4, bufH, N, H1, NCLS, 1);
    init_agg_kernel<<<EW_BLOCKS, TB, 0, stream>>>(bufH, dinv, b4, buf1, N, NCLS);
    scatter_scalar_kernel<<<SCATTER_BLOCKS, TB, 0, stream>>>(ei, enorm, bufH, buf1, E, NCLS);

    // --- log softmax ---
    log_softmax_kernel<<<(N + TB - 1) / TB, TB, 0, stream>>>(buf1, out, N, NCLS);
}